// RMSEnergyExtractor_6854767804675
// MI455X (gfx1250) — compile-verified
//
#include <hip/hip_runtime.h>
#include <hip/hip_bf16.h>

// RMS energy (librosa.feature.rms semantics): frame=2048, hop=512, reflect pad.
// Pass 1: per-hop chunk sums of x^2 via V_WMMA_F32_16X16X4_F32 accumulation
//         (A = ones 16x4, B = squared samples; D columns accumulate partials).
//         One wave per 512-sample chunk; 4x global_load_b128 + 8 WMMAs
//         (two independent accumulators so the XDL pipe can overlap them).
// Pass 2: frame sum = 4 adjacent chunk sums; rms = sqrt(sum / 2048).

#define FRAME  2048
#define HOP    512
#define PAD    1024   // FRAME/2

typedef __attribute__((ext_vector_type(2))) float v2f;
typedef __attribute__((ext_vector_type(8))) float v8f;

__device__ __forceinline__ int reflect_idx(int p, int T) {
    // index into original x for padded position p in [0, T + 2*PAD); T = 2^25 so
    // everything fits in 32-bit int.
    if (p < PAD)      return PAD - p;              // xp[i] = x[PAD - i]
    if (p >= T + PAD) return 2 * T + PAD - 2 - p;  // xp[PAD+T+k] = x[T-2-k]
    return p - PAD;
}

// One wave (32 lanes) per 512-sample chunk of the padded signal.
// Block = 256 threads = 8 waves = 8 chunks.
__global__ void rms_chunk_sums_kernel(const float* __restrict__ x,
                                      float* __restrict__ S,
                                      int nchunks, int T) {
    const int lane  = threadIdx.x & 31;
    const int wave  = threadIdx.x >> 5;
    const int chunk = blockIdx.x * 8 + wave;
    if (chunk >= nchunks) return;              // wave-uniform exit (EXEC stays all-ones)

    v2f ones; ones.x = 1.0f; ones.y = 1.0f;    // A = 16x4 ones matrix (2 VGPRs/lane)
    v8f acc0 = {};                             // two independent f32 accumulators
    v8f acc1 = {};                             // (break WMMA->WMMA D->C serialization)

    const int p0 = chunk * HOP;                // padded range [p0, p0+512)
    const bool interior = (p0 >= PAD) && (p0 + HOP <= T + PAD);  // wave-uniform

    if (interior) {
        // contiguous, coalesced 16B loads straight from x (base 512-float aligned)
        const float4* __restrict__ base =
            reinterpret_cast<const float4*>(x + (p0 - PAD));
        #pragma unroll
        for (int t = 0; t < 4; ++t) {
            float4 v = base[t * 32 + lane];    // 128 samples per iter across the wave
            v2f b0; b0.x = v.x * v.x; b0.y = v.y * v.y;
            v2f b1; b1.x = v.z * v.z; b1.y = v.w * v.w;
            acc0 = __builtin_amdgcn_wmma_f32_16x16x4_f32(
                false, ones, false, b0, (short)0, acc0, false, false);
            acc1 = __builtin_amdgcn_wmma_f32_16x16x4_f32(
                false, ones, false, b1, (short)0, acc1, false, false);
        }
    } else {
        // boundary chunk: reflect-mapped scalar gathers (only 4 chunks total)
        #pragma unroll
        for (int t = 0; t < 4; ++t) {
            int p = p0 + t * 128 + lane * 4;
            float e0 = x[reflect_idx(p,     T)];
            float e1 = x[reflect_idx(p + 1, T)];
            float e2 = x[reflect_idx(p + 2, T)];
            float e3 = x[reflect_idx(p + 3, T)];
            v2f b0; b0.x = e0 * e0; b0.y = e1 * e1;
            v2f b1; b1.x = e2 * e2; b1.y = e3 * e3;
            acc0 = __builtin_amdgcn_wmma_f32_16x16x4_f32(
                false, ones, false, b0, (short)0, acc0, false, false);
            acc1 = __builtin_amdgcn_wmma_f32_16x16x4_f32(
                false, ones, false, b1, (short)0, acc1, false, false);
        }
    }

    // acc[0] on lane L = column-sum for column (L % 16); all 16 rows of D are
    // identical (A's rows are identical), so summing lanes 0..15 of element 0
    // yields the full chunk sum.
    float s = acc0[0] + acc1[0];
    s += __shfl_xor(s, 1);
    s += __shfl_xor(s, 2);
    s += __shfl_xor(s, 4);
    s += __shfl_xor(s, 8);
    if (lane == 0) S[chunk] = s;
}

// out[i] = sqrt( (S[i] + S[i+1] + S[i+2] + S[i+3]) / FRAME )
__global__ void rms_combine_kernel(const float* __restrict__ S,
                                   float* __restrict__ out, int nframes) {
    int i = blockIdx.x * blockDim.x + threadIdx.x;
    if (i >= nframes) return;
    float s = S[i] + S[i + 1] + S[i + 2] + S[i + 3];
    out[i] = sqrtf(s * (1.0f / (float)FRAME));
}

extern "C" void kernel_launch(void* const* d_in, const int* in_sizes, int n_in,
                              void* d_out, int out_size, void* d_ws, size_t ws_size,
                              hipStream_t stream) {
    const float* x = (const float*)d_in[0];
    float* out = (float*)d_out;
    float* S = (float*)d_ws;                   // nchunks floats of scratch

    const int T = in_sizes[0];                 // 33554432
    const int nchunks = T / HOP + 4;           // 65540 hop-chunks of padded signal
    const int nframes = out_size;              // 1 + T/HOP = 65537

    dim3 b1(256), g1((unsigned)((nchunks + 7) / 8)); // 8 waves -> 8 chunks per block
    rms_chunk_sums_kernel<<<g1, b1, 0, stream>>>(x, S, nchunks, T);

    dim3 b2(256), g2((unsigned)((nframes + 255) / 256));
    rms_combine_kernel<<<g2, b2, 0, stream>>>(S, out, nframes);
}